// DrivePostModel_89283780149326
// MI455X (gfx1250) — compile-verified
//
#include <hip/hip_runtime.h>
#include <math.h>

// ---------- problem constants ----------
#define NE_TOT   483840      // total detection entries (all scales, all batches)
#define NE0      368640      // scale0: 32*3*48*80
#define NE1       92160      // scale1: 32*3*24*40
#define NE2       23040      // scale2: 32*3*12*20
#define NPB0      11520      // per-batch entries scale0
#define NPB1       2880
#define NPB2        720
#define NB_SCAN    1890      // NE_TOT / 256 exactly
#define BATCH        32
#define NOBJ        300
#define CONF_T   0.25f
#define IOU_T    0.45f

// output layout (flat float32, concatenated in return order)
#define SZ_BBOX  (BATCH*NOBJ*6)                 // 57600
#define SEG_PIX  (BATCH*384*640)                // 7864320
#define OFF_DA   (SZ_BBOX)
#define OFF_LL   (OFF_DA + SEG_PIX)
#define OFF_COL  (OFF_LL + SEG_PIX)
#define COL_N4   (BATCH*720*1280*3/4)           // 22118400 float4s
#define SEG_PAIR (SEG_PIX/2)                    // 3932160 float4 loads per seg

typedef float f4  __attribute__((ext_vector_type(4)));
typedef float f2  __attribute__((ext_vector_type(2)));
typedef float v2f __attribute__((ext_vector_type(2)));
typedef float v8f __attribute__((ext_vector_type(8)));

__constant__ float c_anch[3][3][2] = {
  {{3.f,9.f},{5.f,11.f},{4.f,20.f}},
  {{7.f,18.f},{6.f,39.f},{12.f,31.f}},
  {{19.f,50.f},{38.f,81.f},{68.f,157.f}}};

__device__ __forceinline__ float sigm(float x){ return 1.0f/(1.0f + __expf(-x)); }

__device__ __forceinline__ void map_entry(int e, int& s, int& b, int& a, int& yy, int& xx,
                                          int& nyv, int& nxv, float& stride){
  int local;
  if (e < NE0)            { s=0; local=e;           nyv=48; nxv=80; stride=8.f;  }
  else if (e < NE0+NE1)   { s=1; local=e-NE0;       nyv=24; nxv=40; stride=16.f; }
  else                    { s=2; local=e-NE0-NE1;   nyv=12; nxv=20; stride=32.f; }
  const int npb = (s==0)?NPB0:((s==1)?NPB1:NPB2);
  b = local / npb; int r = local - b*npb;
  int nyx = nyv*nxv;
  a = r / nyx; int rr = r - a*nyx;
  yy = rr / nxv; xx = rr - yy*nxv;
}

// ---------- K0: init b_out table to -1 ----------
__global__ void k_init_bout(float* bout){
  int i = blockIdx.x*256 + threadIdx.x;
  if (i < BATCH*NOBJ*5) bout[i] = -1.0f;
}

// ---------- K1: confidence mask + per-block sums ----------
__global__ __launch_bounds__(256) void k_mask(const float* __restrict__ d0,
                                              const float* __restrict__ d1,
                                              const float* __restrict__ d2,
                                              int* __restrict__ mask, int* __restrict__ bsum){
  int e = blockIdx.x*256 + threadIdx.x;
  int s,b,a,yy,xx,nyv,nxv; float stride;
  map_entry(e,s,b,a,yy,xx,nyv,nxv,stride);
  const float* d = (s==0)?d0:((s==1)?d1:d2);
  size_t base = ((((size_t)b*3 + a)*nyv + yy)*(size_t)nxv + xx)*6;
  float obj = sigm(d[base+4]);
  float cm  = obj * sigm(d[base+5]);
  int m = (obj > CONF_T) && (cm > CONF_T);
  mask[e] = m;
  __shared__ int red[256];
  red[threadIdx.x] = m; __syncthreads();
  for (int off=128; off>0; off>>=1){
    if (threadIdx.x < off) red[threadIdx.x] += red[threadIdx.x+off];
    __syncthreads();
  }
  if (threadIdx.x==0) bsum[blockIdx.x] = red[0];
}

// ---------- K2: serial exclusive scan of block sums (tiny) ----------
__global__ void k_scan(const int* __restrict__ bsum, int* __restrict__ boff){
  int acc = 0;
  for (int i=0;i<NB_SCAN;++i){ boff[i]=acc; acc+=bsum[i]; }
}

// ---------- K3: decode + global-cumsum scatter ----------
__global__ __launch_bounds__(256) void k_scatter(const float* __restrict__ d0,
                                                 const float* __restrict__ d1,
                                                 const float* __restrict__ d2,
                                                 const int* __restrict__ mask,
                                                 const int* __restrict__ boff,
                                                 float* __restrict__ bout){
  int e = blockIdx.x*256 + threadIdx.x;
  int m = mask[e];
  __shared__ int sc[256];
  sc[threadIdx.x] = m; __syncthreads();
  for (int off=1; off<256; off<<=1){
    int v = (threadIdx.x>=off)? sc[threadIdx.x-off] : 0;
    __syncthreads();
    sc[threadIdx.x] += v;
    __syncthreads();
  }
  int k = boff[blockIdx.x] + sc[threadIdx.x] - m;   // global exclusive prefix
  if (m && k < NOBJ){
    int s,b,a,yy,xx,nyv,nxv; float stride;
    map_entry(e,s,b,a,yy,xx,nyv,nxv,stride);
    const float* d = (s==0)?d0:((s==1)?d1:d2);
    size_t base = ((((size_t)b*3 + a)*nyv + yy)*(size_t)nxv + xx)*6;
    float s0=sigm(d[base+0]), s1=sigm(d[base+1]), s2=sigm(d[base+2]);
    float s3=sigm(d[base+3]), s4=sigm(d[base+4]), s5=sigm(d[base+5]);
    float px=(s0*2.f-0.5f+(float)xx)*stride, py=(s1*2.f-0.5f+(float)yy)*stride;
    float w=(s2*2.f)*(s2*2.f)*c_anch[s][a][0], h=(s3*2.f)*(s3*2.f)*c_anch[s][a][1];
    float* o = bout + ((size_t)b*NOBJ + k)*5;
    o[0]=px-w*0.5f; o[1]=py-h*0.5f; o[2]=px+w*0.5f; o[3]=py+h*0.5f; o[4]=s5*s4;
  }
}

// ---------- K4: per-batch sort + WMMA-tiled IoU + ballot NMS ----------
__global__ __launch_bounds__(256) void k_nms(const float* __restrict__ bout,
                                             const float* __restrict__ ratios,
                                             const float* __restrict__ paddings,
                                             float* __restrict__ out){
  const int bb   = blockIdx.x;
  const int tid  = threadIdx.x;
  const int lane = tid & 31;
  const int wv   = tid >> 5;

  __shared__ float ss[512]; __shared__ int si[512];
  __shared__ float X1[304],Y1[304],X2[304],Y2[304],AR[304],SC[304];
  __shared__ unsigned int SUP[304][10];
  __shared__ unsigned int KEEPW[10];
  __shared__ float OB[300][4]; __shared__ float OS[300];
  __shared__ int TK;

  const float padx = paddings[bb*2+0], pady = paddings[bb*2+1];
  const float rx = 1.0f/ratios[bb*2+0], ry = 1.0f/ratios[bb*2+1];

  // 1) load + transform boxes, fill sort keys
  for (int i=tid; i<512; i+=256){
    if (i < 300){
      const float* r = bout + ((size_t)bb*NOBJ + i)*5;
      X1[i]=(r[0]-padx)*rx; Y1[i]=(r[1]-pady)*ry;
      X2[i]=(r[2]-padx)*rx; Y2[i]=(r[3]-pady)*ry;
      ss[i]=r[4];
    } else {
      ss[i]=-__builtin_inff();
      if (i < 304){ X1[i]=Y1[i]=X2[i]=Y2[i]=0.f; }
    }
    si[i]=i;
  }
  __syncthreads();

  // 2) bitonic sort, descending score, stable via index tiebreak
  for (int k=2; k<=512; k<<=1)
    for (int j=k>>1; j>0; j>>=1){
      for (int t=tid; t<512; t+=256){
        int p = t ^ j;
        if (p > t){
          float a=ss[t], b=ss[p]; int ia=si[t], ib=si[p];
          bool inOrder = (a > b) || (a==b && ia < ib);
          bool dir = ((t & k) == 0);
          if (inOrder != dir){ ss[t]=b; ss[p]=a; si[t]=ib; si[p]=ia; }
        }
      }
      __syncthreads();
    }

  // 3) gather boxes into sorted order
  float gx1[2],gy1[2],gx2[2],gy2[2];
  for (int u=0;u<2;++u){
    int p = tid + u*256;
    if (p < 304){
      int j = si[p];
      if (j < 300){ gx1[u]=X1[j]; gy1[u]=Y1[j]; gx2[u]=X2[j]; gy2[u]=Y2[j]; }
      else        { gx1[u]=gy1[u]=gx2[u]=gy2[u]=0.f; }
    }
  }
  __syncthreads();
  for (int u=0;u<2;++u){
    int p = tid + u*256;
    if (p < 304){
      X1[p]=gx1[u]; Y1[p]=gy1[u]; X2[p]=gx2[u]; Y2[p]=gy2[u];
      AR[p]=(gx2[u]-gx1[u])*(gy2[u]-gy1[u]);
      SC[p]=ss[p];
    }
  }
  // zero suppression bitmask
  for (int w=tid; w<304*10; w+=256) ((unsigned int*)SUP)[w]=0u;
  __syncthreads();

  // 4) IoU tiles: union term a_i + a_j via V_WMMA_F32_16X16X4_F32, inter via VALU.
  //    Division-free test: iou > thr  <=>  union > 0 && inter > thr*union.
  const int half = lane >> 4, n = lane & 15;
  for (int t5=wv; t5<19*19; t5+=8){
    int ti = t5/19, tj = t5%19;
    // A (16x4): col0=areas, col1=ones  | lanes16-31 carry K=2,3 -> zero
    v2f afrag; afrag.x = (half==0)? AR[ti*16+n] : 0.f;
               afrag.y = (half==0)? 1.f         : 0.f;
    // B (4x16): row0=ones, row1=areas | lanes16-31 carry K=2,3 -> zero
    v2f bfrag; bfrag.x = (half==0)? 1.f         : 0.f;
               bfrag.y = (half==0)? AR[tj*16+n] : 0.f;
    v8f acc = {0.f,0.f,0.f,0.f,0.f,0.f,0.f,0.f};
    acc = __builtin_amdgcn_wmma_f32_16x16x4_f32(false, afrag, false, bfrag,
                                                (short)0, acc, false, false);
    #pragma unroll
    for (int r=0; r<8; ++r){
      int i  = ti*16 + r + 8*half;
      int jc = tj*16 + n;
      float cw = fminf(X2[i],X2[jc]) - fmaxf(X1[i],X1[jc]); cw = fmaxf(cw,0.f);
      float ch = fminf(Y2[i],Y2[jc]) - fmaxf(Y1[i],Y1[jc]); ch = fmaxf(ch,0.f);
      float inter = cw*ch;
      float uni = acc[r] - inter;
      bool pred = (i < 300) && (jc < 300) && (uni > 0.f) && (inter > IOU_T*uni);
      unsigned long long bal = __ballot((int)pred);
      unsigned sh = (unsigned)(tj & 1) * 16u;
      if (lane == 0)  atomicOr(&SUP[ti*16+r  ][tj>>1], ((unsigned)(bal      & 0xFFFFu)) << sh);
      if (lane == 16) atomicOr(&SUP[ti*16+r+8][tj>>1], ((unsigned)((bal>>16)& 0xFFFFu)) << sh);
    }
  }
  __syncthreads();

  // 5) sequential keep scan (wave 0), 300 bits across 10 words
  if (tid < 32){
    unsigned kw = 0u;
    if (lane < 9) kw = 0xFFFFFFFFu; else if (lane == 9) kw = 0x00000FFFu;
    for (int i=0;i<300;++i){
      int iw = i >> 5, ib = i & 31;
      unsigned low = (lane < iw) ? 0xFFFFFFFFu
                   : (lane == iw) ? ((ib==0)?0u:((1u<<ib)-1u)) : 0u;
      unsigned t = (lane < 10) ? (SUP[i][lane] & kw & low) : 0u;
      unsigned long long bal = __ballot((int)(t != 0u));
      if (bal != 0ull && lane == iw) kw &= ~(1u << ib);
    }
    if (lane < 10) KEEPW[lane] = kw;
  }
  __syncthreads();
  if (tid == 0){
    int tk=0;
    for (int w2=0; w2<10; ++w2) tk += __popc(KEEPW[w2]);
    TK = tk;
  }
  __syncthreads();

  // 6) stable compaction of kept entries (score-desc order preserved)
  for (int i=tid; i<300; i+=256){
    int iw = i >> 5, ib = i & 31;
    if ((KEEPW[iw] >> ib) & 1u){
      int rank = 0;
      for (int w2=0; w2<iw; ++w2) rank += __popc(KEEPW[w2]);
      rank += __popc(KEEPW[iw] & ((ib==0)?0u:((1u<<ib)-1u)));
      OB[rank][0]=X1[i]; OB[rank][1]=Y1[i]; OB[rank][2]=X2[i]; OB[rank][3]=Y2[i];
      OS[rank]=SC[i];
    }
  }
  __syncthreads();

  // 7) final writes with the reference's (-1 -> inf), (0 -> inf) quirks
  const float INF = __builtin_inff();
  for (int q=tid; q<300; q+=256){
    float bx[4]; float s;
    if (q < TK){ bx[0]=OB[q][0]; bx[1]=OB[q][1]; bx[2]=OB[q][2]; bx[3]=OB[q][3]; s=OS[q]; }
    else       { bx[0]=bx[1]=bx[2]=bx[3]=0.f; s=0.f; }
    float* o = out + ((size_t)bb*NOBJ + q)*6;
    #pragma unroll
    for (int c=0;c<4;++c){
      float v = bx[c];
      v = (v == -1.0f) ? INF : v;
      v = (v ==  0.0f) ? INF : v;
      o[c] = v;
    }
    o[4] = (s == -1.0f) ? INF : s;
    o[5] = 0.0f;
  }
}

// ---------- K5: fused seg argmax masks (b128 NT load -> b64 NT store) ----------
__global__ __launch_bounds__(256) void k_seg(const float* __restrict__ da,
                                             const float* __restrict__ ll,
                                             float* __restrict__ out){
  int g = blockIdx.x*256 + threadIdx.x;            // 0 .. 2*SEG_PAIR-1
  const f4* src; size_t obase; int pair;
  if (g < SEG_PAIR){ src=(const f4*)da; pair=g;          obase = OFF_DA + (size_t)2*pair; }
  else             { src=(const f4*)ll; pair=g-SEG_PAIR; obase = OFF_LL + (size_t)2*pair; }
  f4 v = __builtin_nontemporal_load(src + pair);   // (s0,s1,s0',s1')
  f2 o; o.x = (v.y >= v.x) ? 1.0f : 0.0f;
        o.y = (v.w >= v.z) ? 1.0f : 0.0f;
  __builtin_nontemporal_store(o, (f2*)(out + obase));
}

// ---------- K6: zero-fill color output (NT b128 stores) ----------
__global__ __launch_bounds__(256) void k_zero(float* __restrict__ out){
  size_t i = (size_t)blockIdx.x*256 + threadIdx.x; // 0 .. COL_N4-1
  f4 z = {0.f,0.f,0.f,0.f};
  __builtin_nontemporal_store(z, (f4*)(out + OFF_COL) + i);
}

// ---------- launch ----------
extern "C" void kernel_launch(void* const* d_in, const int* in_sizes, int n_in,
                              void* d_out, int out_size, void* d_ws, size_t ws_size,
                              hipStream_t stream){
  (void)in_sizes; (void)n_in; (void)out_size; (void)ws_size;
  const float* det0 = (const float*)d_in[0];
  const float* det1 = (const float*)d_in[1];
  const float* det2 = (const float*)d_in[2];
  const float* da   = (const float*)d_in[3];
  const float* ll   = (const float*)d_in[4];
  const float* rat  = (const float*)d_in[5];
  const float* pad  = (const float*)d_in[6];
  float* out = (float*)d_out;

  // workspace layout
  int*   mask = (int*)d_ws;
  int*   bsum = mask + NE_TOT;
  int*   boff = bsum + NB_SCAN;
  float* bout = (float*)(boff + NB_SCAN);

  k_init_bout<<<(BATCH*NOBJ*5 + 255)/256, 256, 0, stream>>>(bout);
  k_mask     <<<NB_SCAN, 256, 0, stream>>>(det0, det1, det2, mask, bsum);
  k_scan     <<<1, 1, 0, stream>>>(bsum, boff);
  k_scatter  <<<NB_SCAN, 256, 0, stream>>>(det0, det1, det2, mask, boff, bout);
  k_nms      <<<BATCH, 256, 0, stream>>>(bout, rat, pad, out);
  k_seg      <<<(2*SEG_PAIR)/256, 256, 0, stream>>>(da, ll, out);
  k_zero     <<<COL_N4/256, 256, 0, stream>>>(out);
}